// Retinex_shading_62466004353171
// MI455X (gfx1250) — compile-verified
//
#include <hip/hip_runtime.h>
#include <hip/hip_bf16.h>

typedef __attribute__((ext_vector_type(2))) float v2f;
typedef __attribute__((ext_vector_type(8))) float v8f;

#define W512 512
#define PLANE (512*512)
#define TILE_M 64
#define TILE_N 64
#define TILE_K 16
#define LDA 20   // padded LDS stride for A tile (conflict-free fragment reads)

#if defined(__AMDGCN__) && __has_builtin(__builtin_amdgcn_global_load_async_to_lds_b128) && __has_builtin(__builtin_amdgcn_s_wait_asynccnt)
#define USE_ASYNC_LDS 1
// The builtin's params are pointers to a 128-bit int vector in AS1 (global,
// printed "__device__") and AS3 (LDS, "__shared__") respectively.
typedef int async_b128 __attribute__((vector_size(16)));
typedef __attribute__((address_space(1))) void        gas_void;
typedef __attribute__((address_space(3))) void        lds_void;
typedef __attribute__((address_space(1))) async_b128* g128p;
typedef __attribute__((address_space(3))) async_b128* l128p;
#else
#define USE_ASYNC_LDS 0
#endif

// ---------------------------------------------------------------------------
// Kernel 0: normalization sums for the 3 Gaussian kernels + min/max slot init.
// Reference normalizes over the FULL ksize (91 / 481 / 1501 taps), even taps
// that fall outside the image, so we need the sum over all taps.
// ---------------------------------------------------------------------------
__global__ __launch_bounds__(256) void init_kernel(float* invsum, unsigned* mnmx) {
    const float sig[3] = {15.f, 80.f, 250.f};
    const int   pad[3] = {45, 240, 750};
    int s = blockIdx.x;
    int ksize = 2 * pad[s] + 1;
    float inv2s2 = 1.0f / (2.0f * sig[s] * sig[s]);
    float sum = 0.f;
    for (int t = threadIdx.x; t < ksize; t += blockDim.x) {
        float d = (float)(t - pad[s]);
        sum += expf(-d * d * inv2s2);
    }
    __shared__ float red[256];
    red[threadIdx.x] = sum;
    __syncthreads();
    for (int off = 128; off > 0; off >>= 1) {
        if (threadIdx.x < off) red[threadIdx.x] += red[threadIdx.x + off];
        __syncthreads();
    }
    if (threadIdx.x == 0) invsum[s] = 1.0f / red[0];
    if (s == 0 && threadIdx.x < 3) {
        mnmx[threadIdx.x]     = 0x7F800000u; // +inf bits (min slot)
        mnmx[3 + threadIdx.x] = 0u;          // 0.0 bits (max slot; data >= 0)
    }
}

// ---------------------------------------------------------------------------
// Kernel 1: build banded Gaussian matrices G[s] (512x512, symmetric):
// G[i][j] = exp(-(i-j)^2/(2s^2)) / fullsum   if |i-j| <= pad else 0.
// Row-times-G == zero-padded "same" convolution along that axis.
// ---------------------------------------------------------------------------
__global__ __launch_bounds__(256) void build_g(float* G, const float* invsum) {
    const float sig[3] = {15.f, 80.f, 250.f};
    const int   pad[3] = {45, 240, 750};
    int s = blockIdx.y;
    int flat = blockIdx.x * 256 + threadIdx.x;   // 0 .. 512*512-1
    int i = flat >> 9, j = flat & 511;
    int d = i - j;
    float v = 0.f;
    if (d <= pad[s] && -d <= pad[s]) {
        float inv2s2 = 1.0f / (2.0f * sig[s] * sig[s]);
        v = expf(-(float)(d * d) * inv2s2) * invsum[s];
    }
    G[(size_t)s * PLANE + flat] = v;
}

// ---------------------------------------------------------------------------
// Kernel 2: WMMA fp32 GEMM, C(512x512) = A(512x512) x B(512x512) per plane.
// Block: 256 threads = 8 wave32s; block tile 64x64, K-tile 16.
// Wave w owns a 16x32 slab: rows (w>>1)*16, cols (w&1)*32 -> two 16x16 accs.
// Tiles staged via async global->LDS DMA (ASYNCcnt), double-buffered, so the
// copy of K-tile i+1 overlaps the 8 WMMAs of K-tile i.
// EPILOG=false: C = acc.   EPILOG=true: C = beta*C + w*log(acc + 1).
// ---------------------------------------------------------------------------
template <bool EPILOG>
__global__ __launch_bounds__(256)
void gemm512(const float* __restrict__ A, long aStride,
             const float* __restrict__ B, long bStride,
             float* __restrict__ C, long cStride,
             float beta, float wgt)
{
    const int tid  = threadIdx.x;
    const int lane = tid & 31;
    const int wave = tid >> 5;
    const int m0 = blockIdx.x * TILE_M;
    const int n0 = blockIdx.y * TILE_N;

    const float* Ap = A + (size_t)blockIdx.z * aStride;
    const float* Bp = B + (size_t)blockIdx.z * bStride;
    float*       Cp = C + (size_t)blockIdx.z * cStride;

    const int rw = (wave >> 1) * 16;  // wave row offset inside block tile
    const int cw = (wave & 1) * 32;   // wave col offset inside block tile

    v8f acc0 = {};
    v8f acc1 = {};

    // cooperative tile loaders: one float4 per thread per tile
    const int arow  = tid >> 2;          // 0..63
    const int acol4 = (tid & 3) << 2;    // 0,4,8,12
    const int brow  = tid >> 4;          // 0..15
    const int bcol4 = (tid & 15) << 2;   // 0..60

    const int lr = lane & 15;
    const int hi = lane >> 4;            // 0 or 1 (half-wave)

    // Inner compute for one staged K-tile (8 x v_wmma_f32_16x16x4_f32).
    auto compute = [&](const float* la, const float* lb) {
#pragma unroll
        for (int kk = 0; kk < TILE_K; kk += 4) {
            // A fragment 16x4 (ISA layout: lanes 0-15 hold K=kk,kk+1;
            //                              lanes 16-31 hold K=kk+2,kk+3)
            const float* ap = &la[(rw + lr) * LDA + kk + (hi << 1)];
            v2f a; a.x = ap[0]; a.y = ap[1];
            // B fragments 4x16 (vgpr0 row kk+2*hi, vgpr1 row kk+1+2*hi)
            const float* bp = &lb[(kk + (hi << 1)) * TILE_N + cw + lr];
            v2f b0, b1;
            b0.x = bp[0];            b0.y = bp[TILE_N];
            b1.x = bp[16];           b1.y = bp[TILE_N + 16];
            acc0 = __builtin_amdgcn_wmma_f32_16x16x4_f32(false, a, false, b0,
                                                         (short)0, acc0, false, false);
            acc1 = __builtin_amdgcn_wmma_f32_16x16x4_f32(false, a, false, b1,
                                                         (short)0, acc1, false, false);
        }
    };

#if USE_ASYNC_LDS
    __shared__ float lA[2][TILE_M * LDA];     // 64x16 tiles, padded rows
    __shared__ float lB[2][TILE_K * TILE_N];  // 16x64 tiles

    // issue async DMA of one K-tile into LDS buffer `buf`
    auto issueTile = [&](int buf, int kt) {
        const float* ga = Ap + (size_t)(m0 + arow) * W512 + kt + acol4;
        float*       la = &lA[buf][arow * LDA + acol4];
        __builtin_amdgcn_global_load_async_to_lds_b128(
            (g128p)(gas_void*)ga, (l128p)(lds_void*)la, 0, 0);
        const float* gb = Bp + (size_t)(kt + brow) * W512 + n0 + bcol4;
        float*       lb = &lB[buf][brow * TILE_N + bcol4];
        __builtin_amdgcn_global_load_async_to_lds_b128(
            (g128p)(gas_void*)gb, (l128p)(lds_void*)lb, 0, 0);
    };

    int cur = 0;
    issueTile(0, 0);
    for (int kt = 0; kt < W512; kt += TILE_K) {
        __builtin_amdgcn_s_wait_asynccnt(0);   // our tile landed in LDS
        __syncthreads();                       // everyone's tile landed
        if (kt + TILE_K < W512)
            issueTile(cur ^ 1, kt + TILE_K);   // prefetch next K-tile
        compute(&lA[cur][0], &lB[cur][0]);
        __syncthreads();                       // all waves done reading cur
        cur ^= 1;
    }
#else
    __shared__ float lA1[TILE_M * LDA];
    __shared__ float lB1[TILE_K * TILE_N];
    for (int kt = 0; kt < W512; kt += TILE_K) {
        float4 av = *(const float4*)(Ap + (size_t)(m0 + arow) * W512 + kt + acol4);
        *(float4*)(&lA1[arow * LDA + acol4]) = av;
        float4 bv = *(const float4*)(Bp + (size_t)(kt + brow) * W512 + n0 + bcol4);
        *(float4*)(&lB1[brow * TILE_N + bcol4]) = bv;
        __syncthreads();
        compute(lA1, lB1);
        __syncthreads();
    }
#endif

    // C/D layout: VGPR v -> rows v (lanes 0-15) and v+8 (lanes 16-31), col = lane%16
    const int hi8 = hi << 3;
#pragma unroll
    for (int v = 0; v < 8; ++v) {
        int row = m0 + rw + v + hi8;
        int col = n0 + cw + lr;
        size_t i0 = (size_t)row * W512 + col;
        if (EPILOG) {
            float p0 = 0.f, p1 = 0.f;
            if (beta != 0.0f) { p0 = Cp[i0]; p1 = Cp[i0 + 16]; }
            Cp[i0]      = beta * p0 + wgt * logf(acc0[v] + 1.0f);
            Cp[i0 + 16] = beta * p1 + wgt * logf(acc1[v] + 1.0f);
        } else {
            Cp[i0]      = acc0[v];
            Cp[i0 + 16] = acc1[v];
        }
    }
}

// ---------------------------------------------------------------------------
// Kernel 3: per-channel min/max over (batch, H, W). All values >= 0, so
// uint bit-compare is order-preserving -> atomicMin/Max on reinterpreted bits.
// ---------------------------------------------------------------------------
__global__ __launch_bounds__(256) void minmax_kernel(const float* __restrict__ r,
                                                     unsigned* mnmx) {
    const int c = blockIdx.y;
    const long NPC = 16L * PLANE;
    unsigned mn = 0x7F800000u, mx = 0u;
    for (long i = (long)blockIdx.x * blockDim.x + threadIdx.x; i < NPC;
         i += (long)gridDim.x * blockDim.x) {
        long plane = i >> 18;            // PLANE == 2^18
        long rem   = i & (PLANE - 1);
        unsigned b = __float_as_uint(r[((plane * 3 + c) << 18) + rem]);
        mn = mn < b ? mn : b;
        mx = mx > b ? mx : b;
    }
    __shared__ unsigned smn[256], smx[256];
    smn[threadIdx.x] = mn; smx[threadIdx.x] = mx;
    __syncthreads();
    for (int off = 128; off > 0; off >>= 1) {
        if (threadIdx.x < off) {
            smn[threadIdx.x] = smn[threadIdx.x] < smn[threadIdx.x + off] ? smn[threadIdx.x] : smn[threadIdx.x + off];
            smx[threadIdx.x] = smx[threadIdx.x] > smx[threadIdx.x + off] ? smx[threadIdx.x] : smx[threadIdx.x + off];
        }
        __syncthreads();
    }
    if (threadIdx.x == 0) {
        atomicMin(&mnmx[c], smn[0]);
        atomicMax(&mnmx[3 + c], smx[0]);
    }
}

// ---------------------------------------------------------------------------
// Kernel 4: per-channel min-max stretch, in place.
// ---------------------------------------------------------------------------
__global__ __launch_bounds__(256) void norm_kernel(float* out, const unsigned* mnmx) {
    long i = (long)blockIdx.x * blockDim.x + threadIdx.x;
    const long N = 48L * PLANE;
    if (i >= N) return;
    int c = (int)((i >> 18) % 3);
    float mn = __uint_as_float(mnmx[c]);
    float mx = __uint_as_float(mnmx[3 + c]);
    out[i] = (out[i] - mn) / (mx - mn);
}

// ---------------------------------------------------------------------------
extern "C" void kernel_launch(void* const* d_in, const int* in_sizes, int n_in,
                              void* d_out, int out_size, void* d_ws, size_t ws_size,
                              hipStream_t stream) {
    (void)in_sizes; (void)n_in; (void)out_size;
    const float* img = (const float*)d_in[0];
    float* out = (float*)d_out;
    char* ws = (char*)d_ws;

    float*    invsum = (float*)ws;           // 12 B
    unsigned* mnmx   = (unsigned*)(ws + 16); // 24 B
    float*    G      = (float*)(ws + 256);   // 3 x 1 MB Gaussian matrices
    const size_t tmpOff = 256 + 3 * (size_t)PLANE * sizeof(float);
    float*    tmp    = (float*)(ws + tmpOff);

    // how many planes of intermediate fit in the remaining workspace
    long maxChunk = 1;
    if (ws_size > tmpOff + (size_t)PLANE * sizeof(float))
        maxChunk = (long)((ws_size - tmpOff) / ((size_t)PLANE * sizeof(float)));
    if (maxChunk > 48) maxChunk = 48;
    if (maxChunk < 1)  maxChunk = 1;

    init_kernel<<<3, 256, 0, stream>>>(invsum, mnmx);
    build_g<<<dim3(PLANE / 256, 3), 256, 0, stream>>>(G, invsum);

    const float wgt = 1.0f / 3.0f;
    for (int s = 0; s < 3; ++s) {
        const float* Gs = G + (size_t)s * PLANE;
        float beta = (s == 0) ? 0.0f : 1.0f;
        for (long base = 0; base < 48; base += maxChunk) {
            long n = 48 - base;
            if (n > maxChunk) n = maxChunk;
            dim3 grid(W512 / TILE_M, W512 / TILE_N, (unsigned)n);
            // vertical blur: tmp[p] = G x X[p]
            gemm512<false><<<grid, 256, 0, stream>>>(
                Gs, 0L, img + (size_t)base * PLANE, (long)PLANE,
                tmp, (long)PLANE, 0.f, 0.f);
            // horizontal blur + log epilogue: out[p] = beta*out[p] + w*log(tmp[p] x G + 1)
            gemm512<true><<<grid, 256, 0, stream>>>(
                tmp, (long)PLANE, Gs, 0L,
                out + (size_t)base * PLANE, (long)PLANE, beta, wgt);
        }
    }

    minmax_kernel<<<dim3(512, 3), 256, 0, stream>>>(out, mnmx);
    long N = 48L * PLANE;
    norm_kernel<<<(unsigned)((N + 255) / 256), 256, 0, stream>>>(out, mnmx);
}